// LorenzSDE_11227044511912
// MI455X (gfx1250) — compile-verified
//
#include <hip/hip_runtime.h>

// ---------------------------------------------------------------------------
// Lorenz ShARK SDE integrator, MI455X (gfx1250).
// Memory-bound streaming problem (576 MB @ 23.3 TB/s ~= 25 us floor).
// CDNA5 path used: Tensor Data Mover (tensor_load_to_lds, 6-arg clang-23
// builtin) stages each workgroup's contiguous 15,360 B dW / dH chunks into
// LDS; wave32 f32 VALU does the 5 ShARK steps; conflict-free LDS reads
// (15 dwords/thread, gcd(15,64)=1); NT-hinted global loads/stores.
// ---------------------------------------------------------------------------

typedef __attribute__((ext_vector_type(4))) unsigned int v4u;
typedef __attribute__((ext_vector_type(8))) int          v8i;
typedef __attribute__((ext_vector_type(4))) int          v4i;

#define PPB        256                 // particles per block (8 waves, wave32)
#define TILE_DW    (PPB * 15)          // 3840 dwords = 15360 bytes per array

__device__ __forceinline__ void lorenz_drift(float x, float y, float z,
                                             float& fx, float& fy, float& fz) {
    fx = 10.0f * (y - x);
    fy = x * (28.0f - z) - y;
    fz = x * y - (8.0f / 3.0f) * z;
}

// Build TDM descriptor group0 for a 1-D dword copy into LDS.
__device__ __forceinline__ v4u tdm_group0(unsigned int lds_addr,
                                          unsigned long long gaddr) {
    v4u g0;
    g0.x = 1u;                                         // count=1 (valid), user mode
    g0.y = lds_addr;                                   // LDS byte address
    g0.z = (unsigned int)(gaddr & 0xFFFFFFFFull);      // global_addr[31:0]
    g0.w = (unsigned int)((gaddr >> 32) & 0x01FFFFFFull) | (2u << 30); // [56:32] | type=2
    return g0;
}

// Group1 for: data_size=4B, 1-D tensor of `dim0` dwords, tile of TILE_DW dwords.
__device__ __forceinline__ v8i tdm_group1(unsigned int dim0) {
    v8i g1;
    g1[0] = (int)(2u << 16);                                   // data_size=2 (4B), mask=0, no flags
    g1[1] = (int)((dim0 & 0xFFFFu) << 16);                     // tensor_dim0[15:0]
    g1[2] = (int)(((dim0 >> 16) & 0xFFFFu) | (1u << 16));      // tensor_dim0[31:16] | tensor_dim1=1
    g1[3] = (int)((unsigned)TILE_DW << 16);                    // tensor_dim1 hi=0 | tile_dim0=3840
    g1[4] = 1;                                                 // tile_dim1=1, tile_dim2=0
    g1[5] = (int)((unsigned)TILE_DW);                          // tensor_dim0_stride lo
    g1[6] = 0;                                                 // stride0 hi | stride1 lo
    g1[7] = 0;                                                 // stride1 hi
    return g1;
}

__global__ __launch_bounds__(PPB)
void lorenz_shark_kernel(const float* __restrict__ x,
                         const float* __restrict__ dW,
                         const float* __restrict__ dH,
                         float* __restrict__ out,
                         int n) {
    __shared__ __align__(128) float lw[TILE_DW];
    __shared__ __align__(128) float lh[TILE_DW];

    const int       t        = (int)threadIdx.x;
    const long long blk      = (long long)blockIdx.x;
    const long long tile_off = blk * (long long)TILE_DW;       // dword offset of tile
    const long long total_dw = (long long)n * 15ll;

    if (t == 0) {
        long long    rem  = total_dw - tile_off;
        unsigned int dim0 = (rem >= (long long)TILE_DW) ? (unsigned)TILE_DW
                                                        : (unsigned)rem;   // OOB reads -> 0
        unsigned long long gaW =
            (unsigned long long)(const void*)dW + (unsigned long long)tile_off * 4ull;
        unsigned long long gaH =
            (unsigned long long)(const void*)dH + (unsigned long long)tile_off * 4ull;
        // LDS byte offset = low 32 bits of the flat shared-aperture address.
        unsigned int ldsW = (unsigned int)(unsigned long long)(const void*)lw;
        unsigned int ldsH = (unsigned int)(unsigned long long)(const void*)lh;

        v8i g1 = tdm_group1(dim0);
        v4i g2;  g2[0] = 1; g2[1] = 1; g2[2] = 0; g2[3] = 0;   // dim2=1, dim3=1
        v4i g3;  g3[0] = 0; g3[1] = (int)(1u << 16); g3[2] = 0; g3[3] = 0; // dim4=1
        v8i g4;  g4[0] = 0; g4[1] = 0; g4[2] = 0; g4[3] = 0;   // VADDR4 slot: unused -> zeros
                 g4[4] = 0; g4[5] = 0; g4[6] = 0; g4[7] = 0;

        __builtin_amdgcn_tensor_load_to_lds(tdm_group0(ldsW, gaW), g1, g2, g3, g4, 0);
        __builtin_amdgcn_tensor_load_to_lds(tdm_group0(ldsH, gaH), g1, g2, g3, g4, 0);
        __builtin_amdgcn_s_wait_tensorcnt(0);
    }
    __syncthreads();

    const int p = (int)(blk * PPB) + t;
    if (p < n) {
        // Constants (fold at compile time).
        const float dt   = 0.01f;
        const float cw   = 2.0f * __builtin_sqrtf(0.01f);          // NOISE*sqrt(dt)
        const float ch   = 2.0f * __builtin_sqrtf(0.01f / 12.0f);  // NOISE*sqrt(dt/12)
        const float c56  = 5.0f / 6.0f;
        const float c56h = c56 * dt;

        float yx = __builtin_nontemporal_load(x + 3 * p + 0);
        float yy = __builtin_nontemporal_load(x + 3 * p + 1);
        float yz = __builtin_nontemporal_load(x + 3 * p + 2);

        const int base = t * 15;   // 15 dwords/thread; gcd(15,64)=1 -> bank-conflict-free
#pragma unroll
        for (int k = 0; k < 5; ++k) {
            const float wx = lw[base + 3 * k + 0] * cw;
            const float wy = lw[base + 3 * k + 1] * cw;
            const float wz = lw[base + 3 * k + 2] * cw;
            const float hx = lh[base + 3 * k + 0] * ch;
            const float hy = lh[base + 3 * k + 1] * ch;
            const float hz = lh[base + 3 * k + 2] * ch;

            // z1 = y + g*H
            const float z1x = yx + hx, z1y = yy + hy, z1z = yz + hz;
            float f1x, f1y, f1z;
            lorenz_drift(z1x, z1y, z1z, f1x, f1y, f1z);

            // z2 = y + (5/6) h f1 + (5/6) gW + gH
            const float z2x = yx + c56h * f1x + c56 * wx + hx;
            const float z2y = yy + c56h * f1y + c56 * wy + hy;
            const float z2z = yz + c56h * f1z + c56 * wz + hz;
            float f2x, f2y, f2z;
            lorenz_drift(z2x, z2y, z2z, f2x, f2y, f2z);

            // y1 = y + h (0.4 f1 + 0.6 f2) + gW
            yx = yx + dt * (0.4f * f1x + 0.6f * f2x) + wx;
            yy = yy + dt * (0.4f * f1y + 0.6f * f2y) + wy;
            yz = yz + dt * (0.4f * f1z + 0.6f * f2z) + wz;
        }

        __builtin_nontemporal_store(yx, out + 3 * p + 0);
        __builtin_nontemporal_store(yy, out + 3 * p + 1);
        __builtin_nontemporal_store(yz, out + 3 * p + 2);
    }
}

extern "C" void kernel_launch(void* const* d_in, const int* in_sizes, int n_in,
                              void* d_out, int out_size, void* d_ws, size_t ws_size,
                              hipStream_t stream) {
    const float* x  = (const float*)d_in[0];   // [B,3]
    const float* dW = (const float*)d_in[1];   // [B,5,3]
    const float* dH = (const float*)d_in[2];   // [B,5,3]
    float*       y  = (float*)d_out;           // [B,3]

    const int n      = in_sizes[0] / 3;
    const int blocks = (n + PPB - 1) / PPB;
    hipLaunchKernelGGL(lorenz_shark_kernel, dim3(blocks), dim3(PPB), 0, stream,
                       x, dW, dH, y, n);
}